// LocalAttention_73735998538515
// MI455X (gfx1250) — compile-verified
//
#include <hip/hip_runtime.h>
#include <hip/hip_bf16.h>

// ---------------------------------------------------------------------------
// LocalAttention for MI455X (gfx1250, wave32, WMMA + TDM)
//
// Shapes: x(32,256,4000) f32, W_qkv(256,768), W_out(256,256), b_out(256),
//         pos(9,9).  P=5, HEADS=8, DH=32.  Outputs: out(32,256,4000) then
//         attn(25600,8,5,5), concatenated in d_out.
//
// Block = 1 batch x 16 windows (80 h-positions), 320 threads = 10 wave32.
// Pipeline: TDM x-tile + weight chunks -> fp32 WMMA GEMM1 -> VALU attention
//           -> fp32 WMMA GEMM2 (+bias) -> transposed writeback.
// ---------------------------------------------------------------------------

typedef float v2f __attribute__((ext_vector_type(2)));
typedef float v8f __attribute__((ext_vector_type(8)));
typedef unsigned int v4u __attribute__((ext_vector_type(4)));
typedef int v4i __attribute__((ext_vector_type(4)));
typedef int v8i __attribute__((ext_vector_type(8)));

#define NBATCH   32
#define NCH      256
#define NH       4000
#define NWIN     800
#define WCNT     16          // windows per block
#define HT       80          // h positions per block (= 5 M-tiles of 16)
#define NBLKH    50          // h-chunks per batch
#define NTHREADS 320         // 10 wave32
#define SCALE_F  0.17677669529663687f   // 32^-0.5

// LDS layout (bytes); 283136 <= 320KB WGP LDS
#define XT_OFF    0          // [256][80] f32 x-tile (K-major); reused as AV
#define WB0_OFF   81920      // [256][32] f32 weight chunk, buffer 0
#define WB1_OFF   114688     // [256][32] f32 weight chunk, buffer 1
#define QKV_OFF   147456     // [80][768] f16 qkv; reused as O [80][257] f32
#define DOTS_OFF  270336     // [16][8][5][5] f32
#define SMEM_BYTES 283136

// ---- Tensor Data Mover: 2D tile (f32) global -> LDS ------------------------
// D# per CDNA5 ISA: group0 {count, lds_addr, global_addr, type=2},
// group1 {data_size=4B, tensor_dim0/1, tile_dim0/1, tensor_dim0_stride},
// groups 2/3 zero (tile_dim2..4 = 0 -> unused, 2D tensor).
__device__ __forceinline__ void tdm_load_2d(unsigned lds_off, const void* gptr,
                                            unsigned td0, unsigned td1,
                                            unsigned tile0, unsigned tile1,
                                            unsigned long long stride0)
{
  unsigned long long ga = (unsigned long long)gptr;
  v4u g0;
  g0[0] = 1u;                                      // count=1, user descriptor
  g0[1] = lds_off;                                 // LDS byte address
  g0[2] = (unsigned)(ga & 0xFFFFFFFFull);          // global_addr[31:0]
  g0[3] = (unsigned)((ga >> 32) & 0x1FFFFFFull)    // global_addr[56:32]
        | (2u << 30);                              // type=2 ("image")
  v8i g1;
  g1[0] = (int)(2u << 16);                         // data_size = 4 bytes
  g1[1] = (int)((td0 & 0xFFFFu) << 16);            // tensor_dim0[15:0]
  g1[2] = (int)(((td0 >> 16) & 0xFFFFu) | ((td1 & 0xFFFFu) << 16));
  g1[3] = (int)(((td1 >> 16) & 0xFFFFu) | ((tile0 & 0xFFFFu) << 16));
  g1[4] = (int)(tile1 & 0xFFFFu);                  // tile_dim1; tile_dim2=0
  g1[5] = (int)(unsigned)(stride0 & 0xFFFFFFFFull);
  g1[6] = (int)(unsigned)((stride0 >> 32) & 0xFFFFull); // stride1=0
  g1[7] = 0;
  v4i gz4 = {0, 0, 0, 0};
  v8i gz8 = {0, 0, 0, 0, 0, 0, 0, 0};
  // 6-arg form (clang-23 / therock): (g0, g1, g2, g3, g4?, cpol)
  __builtin_amdgcn_tensor_load_to_lds(g0, g1, gz4, gz4, gz8, 0);
}

__global__ __launch_bounds__(NTHREADS)
void local_attn_kernel(const float* __restrict__ x,
                       const float* __restrict__ Wqkv,
                       const float* __restrict__ Wout,
                       const float* __restrict__ bout,
                       const float* __restrict__ pos,
                       float* __restrict__ out,
                       float* __restrict__ attn)
{
  __shared__ __align__(16) unsigned char smem[SMEM_BYTES];
  float*    xt    = (float*)(smem + XT_OFF);     // [256][80]  K-major A matrix
  float*    wbuf0 = (float*)(smem + WB0_OFF);    // [256][32]
  float*    wbuf1 = (float*)(smem + WB1_OFF);    // [256][32]
  _Float16* qkv   = (_Float16*)(smem + QKV_OFF); // [80][768]
  float*    ovals = (float*)(smem + QKV_OFF);    // [80][257]  (reuse)
  float*    dots  = (float*)(smem + DOTS_OFF);   // [16][8][5][5]

  const int tid  = threadIdx.x;
  const int lane = tid & 31;
  const int wid  = tid >> 5;
  const int b    = blockIdx.x / NBLKH;
  const int cblk = blockIdx.x % NBLKH;
  const int h0   = cblk * HT;
  const int w0   = cblk * WCNT;
  const unsigned lds_base = (unsigned)(unsigned long long)(void*)smem;

  // --- kick off DMA: x-tile (256 rows x 80 cols) + first W_qkv chunk -------
  if (wid == 0) {
    tdm_load_2d(lds_base + XT_OFF, x + ((size_t)b * NCH * NH + h0),
                NH, NCH, HT, NCH, (unsigned long long)NH);
    tdm_load_2d(lds_base + WB0_OFF, Wqkv,
                768u, NCH, 32u, NCH, 768ull);
  }

  // per-wave WMMA tile assignment: mt in [0,5), nt in [0,2)
  const int mt   = wid % 5;
  const int nt   = wid / 5;
  const int mrow = mt * 16 + (lane & 15);
  const int ncol = nt * 16 + (lane & 15);
  const int koff = (lane >> 4) << 1;   // lanes 16-31 hold K+2,K+3
  const int rsel = (lane >> 4) << 3;   // lanes 16-31 hold M rows +8

  // ======================= GEMM1: qkv = Xw @ W_qkv =========================
  for (int c = 0; c < 24; ++c) {
    if (wid == 0) {
      if (c + 1 < 24) {
        tdm_load_2d(lds_base + (((c + 1) & 1) ? WB1_OFF : WB0_OFF),
                    Wqkv + (c + 1) * 32, 768u, NCH, 32u, NCH, 768ull);
        __builtin_amdgcn_s_wait_tensorcnt(1);
      } else {
        __builtin_amdgcn_s_wait_tensorcnt(0);
      }
    }
    __syncthreads();
    const float* wp = (c & 1) ? wbuf1 : wbuf0;

    v8f acc0 = {0, 0, 0, 0, 0, 0, 0, 0};
    v8f acc1 = {0, 0, 0, 0, 0, 0, 0, 0};
#pragma unroll 4
    for (int k = 0; k < 256; k += 8) {
      v2f a0, b0, a1, b1;
      a0.x = xt[(k + koff) * 80 + mrow];
      a0.y = xt[(k + koff + 1) * 80 + mrow];
      b0.x = wp[(k + koff) * 32 + ncol];
      b0.y = wp[(k + koff + 1) * 32 + ncol];
      acc0 = __builtin_amdgcn_wmma_f32_16x16x4_f32(false, a0, false, b0,
                                                   (short)0, acc0, false, false);
      a1.x = xt[(k + 4 + koff) * 80 + mrow];
      a1.y = xt[(k + 5 + koff) * 80 + mrow];
      b1.x = wp[(k + 4 + koff) * 32 + ncol];
      b1.y = wp[(k + 5 + koff) * 32 + ncol];
      acc1 = __builtin_amdgcn_wmma_f32_16x16x4_f32(false, a1, false, b1,
                                                   (short)0, acc1, false, false);
    }
    acc0 += acc1;
    const int gn = c * 32 + ncol;
#pragma unroll
    for (int r = 0; r < 8; ++r)
      qkv[(size_t)(mt * 16 + r + rsel) * 768 + gn] = (_Float16)acc0[r];
    __syncthreads();
  }

  // prefetch first W_out chunk; it overlaps the attention phase
  if (wid == 0)
    tdm_load_2d(lds_base + WB0_OFF, Wout, NCH, NCH, 32u, NCH, 256ull);

  // ========================= attention (VALU) ==============================
  float bias[5];
#pragma unroll
  for (int i = 0; i < 5; ++i) bias[i] = pos[(4 - i) * 9 + (5 - i)];

  // dots[w][h][i][j] = scale * q_i . k_j + bias_i
  for (int idx = tid; idx < 16 * 8 * 25; idx += NTHREADS) {
    const int j = idx % 5, i = (idx / 5) % 5, hh = (idx / 25) % 8, w = idx / 200;
    const _Float16* qp = qkv + (size_t)(w * 5 + i) * 768 + hh * 32;
    const _Float16* kp = qkv + (size_t)(w * 5 + j) * 768 + 256 + hh * 32;
    float s = 0.f;
#pragma unroll
    for (int d = 0; d < 32; ++d) s += (float)qp[d] * (float)kp[d];
    dots[idx] = s * SCALE_F + bias[i];
  }
  __syncthreads();

  // softmax rows; write attn probabilities to global + back to LDS
  for (int r = tid; r < 16 * 8 * 5; r += NTHREADS) {
    const int i = r % 5, hh = (r / 5) % 8, w = r / 40;
    float* dp = dots + (((w * 8 + hh) * 5 + i) * 5);
    float m = dp[0];
#pragma unroll
    for (int j = 1; j < 5; ++j) m = fmaxf(m, dp[j]);
    float e[5], sum = 0.f;
#pragma unroll
    for (int j = 0; j < 5; ++j) { e[j] = expf(dp[j] - m); sum += e[j]; }
    const float inv = 1.f / sum;
    const size_t wg = (size_t)b * NWIN + w0 + w;
    float* ap = attn + (((wg * 8 + hh) * 5 + i) * 5);
#pragma unroll
    for (int j = 0; j < 5; ++j) { const float p = e[j] * inv; dp[j] = p; ap[j] = p; }
  }
  __syncthreads();

  // AV = attn @ v, stored K-major into the (dead) x-tile buffer: xt[col][row]
  for (int idx = tid; idx < HT * 256; idx += NTHREADS) {
    const int row = idx % 80, col = idx / 80;
    const int w = row / 5, i = row % 5, hh = col >> 5;
    const float* dp = dots + (((w * 8 + hh) * 5 + i) * 5);
    const _Float16* vp = qkv + (size_t)(w * 5) * 768 + 512 + col;
    float s = 0.f;
#pragma unroll
    for (int j = 0; j < 5; ++j) s += dp[j] * (float)vp[(size_t)j * 768];
    xt[col * 80 + row] = s;
  }
  __syncthreads();

  // ==================== GEMM2: O = AV @ W_out + b_out ======================
  for (int c = 0; c < 8; ++c) {
    if (wid == 0) {
      if (c + 1 < 8) {
        tdm_load_2d(lds_base + (((c + 1) & 1) ? WB1_OFF : WB0_OFF),
                    Wout + (c + 1) * 32, NCH, NCH, 32u, NCH, 256ull);
        __builtin_amdgcn_s_wait_tensorcnt(1);
      } else {
        __builtin_amdgcn_s_wait_tensorcnt(0);
      }
    }
    __syncthreads();
    const float* wp = (c & 1) ? wbuf1 : wbuf0;

    v8f acc0 = {0, 0, 0, 0, 0, 0, 0, 0};
    v8f acc1 = {0, 0, 0, 0, 0, 0, 0, 0};
#pragma unroll 4
    for (int k = 0; k < 256; k += 8) {
      v2f a0, b0, a1, b1;
      a0.x = xt[(k + koff) * 80 + mrow];
      a0.y = xt[(k + koff + 1) * 80 + mrow];
      b0.x = wp[(k + koff) * 32 + ncol];
      b0.y = wp[(k + koff + 1) * 32 + ncol];
      acc0 = __builtin_amdgcn_wmma_f32_16x16x4_f32(false, a0, false, b0,
                                                   (short)0, acc0, false, false);
      a1.x = xt[(k + 4 + koff) * 80 + mrow];
      a1.y = xt[(k + 5 + koff) * 80 + mrow];
      b1.x = wp[(k + 4 + koff) * 32 + ncol];
      b1.y = wp[(k + 5 + koff) * 32 + ncol];
      acc1 = __builtin_amdgcn_wmma_f32_16x16x4_f32(false, a1, false, b1,
                                                   (short)0, acc1, false, false);
    }
    acc0 += acc1;
    const int gn = c * 32 + ncol;
    const float bv = bout[gn];
#pragma unroll
    for (int r = 0; r < 8; ++r)
      ovals[(size_t)(mt * 16 + r + rsel) * 257 + gn] = acc0[r] + bv;
    __syncthreads();
  }

  // ===================== transposed writeback to out =======================
  for (int idx = tid; idx < HT * 256; idx += NTHREADS) {
    const int j = idx % 80, ch = idx / 80;
    out[(size_t)b * NCH * NH + (size_t)ch * NH + h0 + j] = ovals[j * 257 + ch];
  }
}

extern "C" void kernel_launch(void* const* d_in, const int* in_sizes, int n_in,
                              void* d_out, int out_size, void* d_ws, size_t ws_size,
                              hipStream_t stream) {
  const float* x    = (const float*)d_in[0];
  const float* Wqkv = (const float*)d_in[1];
  const float* Wout = (const float*)d_in[2];
  const float* bout = (const float*)d_in[3];
  const float* pos  = (const float*)d_in[4];
  float* out  = (float*)d_out;
  float* attn = out + (size_t)NBATCH * NCH * NH;   // second tuple element
  (void)in_sizes; (void)n_in; (void)out_size; (void)d_ws; (void)ws_size;

  local_attn_kernel<<<dim3(NBATCH * NBLKH), dim3(NTHREADS), 0, stream>>>(
      x, Wqkv, Wout, bout, pos, out, attn);
}